// GStarModel_32890859552794
// MI455X (gfx1250) — compile-verified
//
#include <hip/hip_runtime.h>

typedef float v2f __attribute__((ext_vector_type(2)));
typedef float v8f __attribute__((ext_vector_type(8)));

#define D 64

__global__ void zero_f32_kernel(float* __restrict__ p, int n) {
  int i = blockIdx.x * blockDim.x + threadIdx.x;
  if (i < n) p[i] = 0.0f;
}

__global__ void deg_kernel(const int* __restrict__ dst, const float* __restrict__ ew,
                           float* __restrict__ deg, int E) {
  int e = blockIdx.x * blockDim.x + threadIdx.x;
  if (e < E) atomicAdd(&deg[dst[e]], ew[e]);
}

__global__ void dinv_kernel(const float* __restrict__ deg, float* __restrict__ dinv, int N) {
  int i = blockIdx.x * blockDim.x + threadIdx.x;
  if (i < N) {
    float d = deg[i] + 1.0f;  // +1 self-loop weight; always > 0
    dinv[i] = rsqrtf(d);
  }
}

__global__ void norm_kernel(const int* __restrict__ src, const int* __restrict__ dst,
                            const float* __restrict__ ew, const float* __restrict__ dinv,
                            float* __restrict__ nrm, int E, int N) {
  int e = blockIdx.x * blockDim.x + threadIdx.x;
  if (e < E) {
    nrm[e] = dinv[src[e]] * ew[e] * dinv[dst[e]];
  } else if (e < E + N) {
    float di = dinv[e - E];
    nrm[e] = di * di;  // self-loop: dinv * 1 * dinv
  }
}

__global__ void bias_init_kernel(float* __restrict__ h, const float* __restrict__ b, int N) {
  int i = blockIdx.x * blockDim.x + threadIdx.x;
  if (i < N * D) h[i] = b[i & (D - 1)];
}

// Y[nrows x 64] = act(X[nrows x 64]) @ W[64 x 64]  via V_WMMA_F32_16X16X4_F32.
// One wave -> one 16x16 tile. 8 waves / 256-thread block. RELU is compile-time
// so the relu layers emit a single v_max per A element (no cndmask selects).
template <bool RELU>
__global__ void gemm64_wmma_kernel(const float* __restrict__ X, const float* __restrict__ W,
                                   float* __restrict__ Y, int nrows) {
  int wave = threadIdx.x >> 5;
  int lane = threadIdx.x & 31;
  int tile = blockIdx.x * (blockDim.x >> 5) + wave;  // wave-uniform
  int rowTile = tile >> 2;
  int colTile = tile & 3;
  if (rowTile * 16 >= nrows) return;  // whole-wave exit; EXEC stays all-ones for WMMA
  int half = lane >> 4;   // 0: lanes 0-15, 1: lanes 16-31
  int mn = lane & 15;     // M for A-loads, N for B-loads
  const float* xrow = X + (size_t)(rowTile * 16 + mn) * D;
  const float* wcol = W + colTile * 16 + mn;
  v8f c = {};
#pragma unroll
  for (int k = 0; k < D; k += 4) {
    // A 16x4 f32: VGPR v, lane-half h -> K = k + v + 2h  (contiguous float2 per lane)
    v2f a, b;
    a[0] = xrow[k + 2 * half + 0];
    a[1] = xrow[k + 2 * half + 1];
    if (RELU) { a[0] = fmaxf(a[0], 0.0f); a[1] = fmaxf(a[1], 0.0f); }
    // B 4x16 f32: VGPR v, lane-half h -> K = k + v + 2h, N = mn
    b[0] = wcol[(size_t)(k + 2 * half + 0) * D];
    b[1] = wcol[(size_t)(k + 2 * half + 1) * D];
    c = __builtin_amdgcn_wmma_f32_16x16x4_f32(false, a, false, b, (short)0, c, false, false);
  }
  // C/D 16x16 f32: VGPR v -> M = v + 8*half, N = mn
  float* yout = Y + (size_t)(rowTile * 16 + 8 * half) * D + colTile * 16 + mn;
#pragma unroll
  for (int v = 0; v < 8; ++v) yout[(size_t)v * D] = c[v];
}

// h[dst[e]] += xw[src[e]] * norm[e]  over E real edges + N self-loops.
// 16 threads per edge, float4 per thread -> 4 atomic f32 adds (resolve in L2).
__global__ void scatter_kernel(const int* __restrict__ src, const int* __restrict__ dst,
                               const float* __restrict__ nrm, const float* __restrict__ xw,
                               float* __restrict__ h, int E, int N) {
  long long idx = (long long)blockIdx.x * blockDim.x + threadIdx.x;
  int e = (int)(idx >> 4);
  if (e >= E + N) return;
  int f = ((int)idx & 15) * 4;
  int s, d;
  if (e < E) { s = src[e]; d = dst[e]; } else { s = e - E; d = s; }
  float nm = nrm[e];
  const float4 v = *reinterpret_cast<const float4*>(xw + (size_t)s * D + f);
  float* hp = h + (size_t)d * D + f;
  atomicAdd(hp + 0, v.x * nm);
  atomicAdd(hp + 1, v.y * nm);
  atomicAdd(hp + 2, v.z * nm);
  atomicAdd(hp + 3, v.w * nm);
}

__global__ void pool_kernel(const float* __restrict__ h, const int* __restrict__ batch,
                            float* __restrict__ sums, float* __restrict__ cnt, int N) {
  long long idx = (long long)blockIdx.x * blockDim.x + threadIdx.x;
  int n = (int)(idx >> 4);
  if (n >= N) return;
  int f = ((int)idx & 15) * 4;
  int g = batch[n];
  const float4 v = *reinterpret_cast<const float4*>(h + (size_t)n * D + f);
  float* sp = sums + (size_t)g * D + f;
  atomicAdd(sp + 0, v.x);
  atomicAdd(sp + 1, v.y);
  atomicAdd(sp + 2, v.z);
  atomicAdd(sp + 3, v.w);
  if (f == 0) atomicAdd(&cnt[g], 1.0f);
}

__global__ void final_kernel(const float* __restrict__ sums, const float* __restrict__ cnt,
                             const float* __restrict__ Wl, const float* __restrict__ bl,
                             float* __restrict__ out, int G, int C) {
  int i = blockIdx.x * blockDim.x + threadIdx.x;
  if (i >= G * C) return;
  int g = i / C, c = i % C;
  float inv = 1.0f / fmaxf(cnt[g], 1.0f);
  float acc = bl[c];
#pragma unroll 8
  for (int f = 0; f < D; ++f) acc += sums[(size_t)g * D + f] * inv * Wl[(size_t)f * C + c];
  out[i] = acc;
}

extern "C" void kernel_launch(void* const* d_in, const int* in_sizes, int n_in,
                              void* d_out, int out_size, void* d_ws, size_t ws_size,
                              hipStream_t stream) {
  const float* x   = (const float*)d_in[0];
  const int*   ei  = (const int*)d_in[1];
  const int*   bat = (const int*)d_in[2];
  const float* ew  = (const float*)d_in[3];
  const float* W1  = (const float*)d_in[4];
  const float* b1  = (const float*)d_in[5];
  const float* W2  = (const float*)d_in[6];
  const float* b2  = (const float*)d_in[7];
  const float* W3  = (const float*)d_in[8];
  const float* b3  = (const float*)d_in[9];
  const float* Wl  = (const float*)d_in[10];
  const float* bl  = (const float*)d_in[11];
  float* out = (float*)d_out;

  const int N = in_sizes[0] / D;  // 50000
  const int E = in_sizes[3];      // 800000
  const int C = 10;
  const int G = out_size / C;     // 128

  const int* src = ei;
  const int* dst = ei + E;

  float* ws   = (float*)d_ws;
  float* deg  = ws; ws += N;
  float* dinv = ws; ws += N;
  float* nrm  = ws; ws += (E + N);
  float* xw   = ws; ws += (size_t)N * D;
  float* h    = ws; ws += (size_t)N * D;
  float* sums = ws; ws += (size_t)G * D;  // cnt directly follows sums
  float* cnt  = ws; ws += G;

  auto cdiv = [](long long a, int b) { return (int)((a + b - 1) / b); };

  // Graph-static normalization (fresh every call; deterministic work).
  zero_f32_kernel<<<cdiv(N, 256), 256, 0, stream>>>(deg, N);
  deg_kernel<<<cdiv(E, 256), 256, 0, stream>>>(dst, ew, deg, E);
  dinv_kernel<<<cdiv(N, 256), 256, 0, stream>>>(deg, dinv, N);
  norm_kernel<<<cdiv((long long)E + N, 256), 256, 0, stream>>>(src, dst, ew, dinv, nrm, E, N);

  const int tiles = cdiv(N, 16) * 4;          // 3125 * 4
  const int gemmBlocks = cdiv(tiles, 8);      // 8 waves per 256-thread block
  const int ndGrid = cdiv((long long)N * D, 256);
  const int scGrid = cdiv(((long long)E + N) * 16, 256);

  // Layer 1: xw = x @ W1 ; h = b1 ; h += scatter(xw)
  gemm64_wmma_kernel<false><<<gemmBlocks, 256, 0, stream>>>(x, W1, xw, N);
  bias_init_kernel<<<ndGrid, 256, 0, stream>>>(h, b1, N);
  scatter_kernel<<<scGrid, 256, 0, stream>>>(src, dst, nrm, xw, h, E, N);

  // Layer 2: xw = relu(h) @ W2 ; h = b2 ; h += scatter(xw)
  gemm64_wmma_kernel<true><<<gemmBlocks, 256, 0, stream>>>(h, W2, xw, N);
  bias_init_kernel<<<ndGrid, 256, 0, stream>>>(h, b2, N);
  scatter_kernel<<<scGrid, 256, 0, stream>>>(src, dst, nrm, xw, h, E, N);

  // Layer 3 (no relu on output): xw = relu(h) @ W3 ; h = b3 ; h += scatter(xw)
  gemm64_wmma_kernel<true><<<gemmBlocks, 256, 0, stream>>>(h, W3, xw, N);
  bias_init_kernel<<<ndGrid, 256, 0, stream>>>(h, b3, N);
  scatter_kernel<<<scGrid, 256, 0, stream>>>(src, dst, nrm, xw, h, E, N);

  // Mean pool + linear head.
  zero_f32_kernel<<<cdiv((long long)G * D + G, 256), 256, 0, stream>>>(sums, G * D + G);
  pool_kernel<<<cdiv((long long)N * 16, 256), 256, 0, stream>>>(h, bat, sums, cnt, N);
  final_kernel<<<cdiv(G * C, 256), 256, 0, stream>>>(sums, cnt, Wl, bl, out, G, C);
}